// GCN_14396730377002
// MI455X (gfx1250) — compile-verified
//
#include <hip/hip_runtime.h>
#include <hip/hip_bf16.h>

typedef __attribute__((ext_vector_type(2))) float v2f;
typedef __attribute__((ext_vector_type(8))) float v8f;

// ---------------------------------------------------------------------------
// Degree / normalization
// ---------------------------------------------------------------------------
__global__ void k_init_deg(float* __restrict__ deg, int n) {
  int i = blockIdx.x * blockDim.x + threadIdx.x;
  if (i < n) deg[i] = 1.0f;  // self-loop contribution
}

__global__ void k_count_deg(const int* __restrict__ dst, float* __restrict__ deg, int ne) {
  int e = blockIdx.x * blockDim.x + threadIdx.x;
  if (e < ne) unsafeAtomicAdd(&deg[__builtin_nontemporal_load(dst + e)], 1.0f);
}

__global__ void k_dinv(const float* __restrict__ deg, float* __restrict__ dinv, int n) {
  int i = blockIdx.x * blockDim.x + threadIdx.x;
  if (i < n) dinv[i] = rsqrtf(deg[i]);
}

// ---------------------------------------------------------------------------
// N x 16 @ 16 x 16 GEMM via V_WMMA_F32_16X16X4_F32.
// One wave computes a 16x16 output tile (16 node rows), K=16 -> 4 WMMA steps.
// A frag (16x4, MxK):  lane L holds A[M=L&15][kh+0 / kh+1], kh = (L>=16)?2:0
// B frag (4x16, KxN):  lane L holds B[kh+0 / kh+1][N=L&15]
// C (16x16): vgpr r, lane L -> C[(L<16 ? r : 8+r)][L&15]
// ---------------------------------------------------------------------------
template <bool RELU_A>
__global__ void k_gemm_n16(const float* __restrict__ A, const float* __restrict__ B,
                           float* __restrict__ C, int nrows) {
  int wid  = (blockIdx.x * blockDim.x + threadIdx.x) >> 5;
  int lane = threadIdx.x & 31;
  int row0 = wid * 16;
  if (row0 >= nrows) return;                 // uniform per wave
  int m  = lane & 15;
  int kh = (lane >> 4) << 1;                 // 0 or 2
  const float* arow = A + (row0 + m) * 16;
  v8f c = {};
#pragma unroll
  for (int k0 = 0; k0 < 16; k0 += 4) {
    v2f a, b;
    float a0 = arow[k0 + kh + 0];
    float a1 = arow[k0 + kh + 1];
    if (RELU_A) { a0 = fmaxf(a0, 0.0f); a1 = fmaxf(a1, 0.0f); }
    a.x = a0; a.y = a1;
    b.x = B[(k0 + kh + 0) * 16 + m];
    b.y = B[(k0 + kh + 1) * 16 + m];
    c = __builtin_amdgcn_wmma_f32_16x16x4_f32(false, a, false, b, (short)0, c, false, false);
  }
  int rbase = row0 + ((lane >> 4) << 3);
#pragma unroll
  for (int r = 0; r < 8; ++r) C[(rbase + r) * 16 + m] = c[r];
}

// ---------------------------------------------------------------------------
// agg[i*16+f] = hw[i*16+f] * dinv[i]^2 + bias[f]   (self-loop + bias folded in)
// ---------------------------------------------------------------------------
__global__ void k_init_agg(const float* __restrict__ hw, const float* __restrict__ dinv,
                           const float* __restrict__ bias, float* __restrict__ agg, int n16) {
  int i = blockIdx.x * blockDim.x + threadIdx.x;
  if (i >= n16) return;
  int node = i >> 4, f = i & 15;
  float di = dinv[node];
  agg[i] = hw[i] * di * di + bias[f];
}

// ---------------------------------------------------------------------------
// Edge scatter: agg[dst] += hw[src] * dinv[src]*dinv[dst]
// 4 lanes per edge: each lane gathers one float4 (16B) and issues 4 atomics.
// 4x the memory-level parallelism of a 1-thread-per-edge scheme.
// ---------------------------------------------------------------------------
__global__ void k_scatter(const int* __restrict__ src, const int* __restrict__ dst,
                          const float* __restrict__ dinv, const float* __restrict__ hw,
                          float* __restrict__ agg, int ne) {
  int t = blockIdx.x * blockDim.x + threadIdx.x;
  int e = t >> 2, q = t & 3;
  if (e >= ne) return;
  int s = __builtin_nontemporal_load(src + e);   // edge list streamed once: NT
  int d = __builtin_nontemporal_load(dst + e);
  float w = dinv[s] * dinv[d];
  float4 v = ((const float4*)(hw + s * 16))[q];
  float* ad = agg + d * 16 + q * 4;
  unsafeAtomicAdd(ad + 0, v.x * w);
  unsafeAtomicAdd(ad + 1, v.y * w);
  unsafeAtomicAdd(ad + 2, v.z * w);
  unsafeAtomicAdd(ad + 3, v.w * w);
}

// ---------------------------------------------------------------------------
// Split-K GEMM: relu(A[64 x lda]) @ B[lda x ncols] accumulated with f32 atomics
// into acc[64 x ncols]. One wave = (tile_m, tile_n, k-split) triple.
// KLEN compile-time so the k-loop unrolls (no exec-masked loop around WMMA).
// ---------------------------------------------------------------------------
template <int KLEN>
__global__ void k_gemm_splitk_relu(const float* __restrict__ A, int lda,
                                   const float* __restrict__ B, int ncols,
                                   float* __restrict__ acc,
                                   int tiles_n, int nsplit) {
  int wid  = (blockIdx.x * blockDim.x + threadIdx.x) >> 5;
  int lane = threadIdx.x & 31;
  int split = wid % nsplit;
  int tn    = (wid / nsplit) % tiles_n;
  int tm    = wid / (nsplit * tiles_n);
  int m  = lane & 15;
  int kh = (lane >> 4) << 1;
  const float* arow = A + (size_t)(tm * 16 + m) * lda + split * KLEN;
  const float* bcol = B + (size_t)(split * KLEN) * ncols + tn * 16 + m;
  v8f c = {};
#pragma unroll 4
  for (int k0 = 0; k0 < KLEN; k0 += 4) {
    v2f a, b;
    a.x = fmaxf(arow[k0 + kh + 0], 0.0f);
    a.y = fmaxf(arow[k0 + kh + 1], 0.0f);
    b.x = bcol[(size_t)(k0 + kh + 0) * ncols];
    b.y = bcol[(size_t)(k0 + kh + 1) * ncols];
    c = __builtin_amdgcn_wmma_f32_16x16x4_f32(false, a, false, b, (short)0, c, false, false);
  }
  int rbase = tm * 16 + ((lane >> 4) << 3);
#pragma unroll
  for (int r = 0; r < 8; ++r)
    unsafeAtomicAdd(&acc[(rbase + r) * ncols + tn * 16 + m], c[r]);
}

// ---------------------------------------------------------------------------
// out[i] = relu(in[i] + bias[i % ncols])
// ---------------------------------------------------------------------------
__global__ void k_bias_relu(const float* __restrict__ in, const float* __restrict__ bias,
                            float* __restrict__ out, int n, int ncols) {
  int i = blockIdx.x * blockDim.x + threadIdx.x;
  if (i < n) out[i] = fmaxf(in[i] + bias[i % ncols], 0.0f);
}

__global__ void k_zero(float* __restrict__ p, int n) {
  int i = blockIdx.x * blockDim.x + threadIdx.x;
  if (i < n) p[i] = 0.0f;
}

// ---------------------------------------------------------------------------
// X = phid[64x32] @ p2_w[32x1024] + p2_b  -> straight to d_out.
// K=32 -> 8 WMMA; 4 x 64 tiles = 256 waves.
// ---------------------------------------------------------------------------
__global__ void k_gemm_final(const float* __restrict__ A, const float* __restrict__ B,
                             const float* __restrict__ bias, float* __restrict__ out) {
  int wid  = (blockIdx.x * blockDim.x + threadIdx.x) >> 5;
  int lane = threadIdx.x & 31;
  int tm = wid >> 6;
  int tn = wid & 63;
  int m  = lane & 15;
  int kh = (lane >> 4) << 1;
  const float* arow = A + (tm * 16 + m) * 32;
  const float* bcol = B + tn * 16 + m;
  v8f c = {};
#pragma unroll
  for (int k0 = 0; k0 < 32; k0 += 4) {
    v2f a, b;
    a.x = arow[k0 + kh + 0];
    a.y = arow[k0 + kh + 1];
    b.x = bcol[(k0 + kh + 0) * 1024];
    b.y = bcol[(k0 + kh + 1) * 1024];
    c = __builtin_amdgcn_wmma_f32_16x16x4_f32(false, a, false, b, (short)0, c, false, false);
  }
  int rbase = tm * 16 + ((lane >> 4) << 3);
  float bv = bias[tn * 16 + m];
#pragma unroll
  for (int r = 0; r < 8; ++r)
    out[(rbase + r) * 1024 + tn * 16 + m] = c[r] + bv;
}

// V[b] = vhid[b,:16] . v2_w + v2_b
__global__ void k_value_final(const float* __restrict__ vhid, const float* __restrict__ w,
                              const float* __restrict__ b, float* __restrict__ out) {
  int bi = threadIdx.x;  // 64 threads
  float s = 0.0f;
#pragma unroll
  for (int j = 0; j < 16; ++j) s += vhid[bi * 16 + j] * w[j];
  out[bi] = s + b[0];
}

// ---------------------------------------------------------------------------
extern "C" void kernel_launch(void* const* d_in, const int* in_sizes, int n_in,
                              void* d_out, int out_size, void* d_ws, size_t ws_size,
                              hipStream_t stream) {
  const float* x    = (const float*)d_in[0];
  const int*   ei   = (const int*)d_in[1];
  const float* W1   = (const float*)d_in[2];
  const float* b1   = (const float*)d_in[3];
  const float* W2   = (const float*)d_in[4];
  const float* b2   = (const float*)d_in[5];
  const float* p1w  = (const float*)d_in[6];
  const float* p1b  = (const float*)d_in[7];
  const float* p2w  = (const float*)d_in[8];
  const float* p2b  = (const float*)d_in[9];
  const float* v1w  = (const float*)d_in[10];
  const float* v1b  = (const float*)d_in[11];
  const float* v2w  = (const float*)d_in[12];
  const float* v2b  = (const float*)d_in[13];

  const int N  = in_sizes[0] / 16;   // 65536 nodes
  const int E  = in_sizes[1] / 2;    // 2097152 edges
  const int* esrc = ei;
  const int* edst = ei + E;

  float* ws   = (float*)d_ws;
  float* deg  = ws;                 // N
  float* dinv = deg + N;            // N
  float* bufA = dinv + N;           // 16N  (hw of current layer)
  float* bufB = bufA + (size_t)16 * N;  // 16N  (aggregate of current layer)
  float* pacc = bufB + (size_t)16 * N;  // 64*32
  float* vacc = pacc + 64 * 32;         // 64*16
  float* phid = vacc + 64 * 16;         // 64*32
  float* vhid = phid + 64 * 32;         // 64*16

  float* Xout = (float*)d_out;           // [64,1024]
  float* Vout = Xout + 64 * 1024;        // [64,1]

  const int B256 = 256;
  // --- normalization ---
  k_init_deg<<<N / B256, B256, 0, stream>>>(deg, N);
  k_count_deg<<<E / B256, B256, 0, stream>>>(edst, deg, E);
  k_dinv<<<N / B256, B256, 0, stream>>>(deg, dinv, N);

  // --- GCN layer 1: hw1 = x @ W1 ; agg1 = self-loop + b1 + edge scatter ---
  int gemm_blocks    = (N / 16) / 8;   // 8 waves per 256-thread block
  int scatter_blocks = (4 * E) / B256; // 4 lanes per edge
  k_gemm_n16<false><<<gemm_blocks, B256, 0, stream>>>(x, W1, bufA, N);
  k_init_agg<<<(16 * N) / B256, B256, 0, stream>>>(bufA, dinv, b1, bufB, 16 * N);
  k_scatter<<<scatter_blocks, B256, 0, stream>>>(esrc, edst, dinv, bufA, bufB, E);

  // --- GCN layer 2: hw2 = relu(agg1) @ W2 ; agg2 likewise ---
  k_gemm_n16<true><<<gemm_blocks, B256, 0, stream>>>(bufB, W2, bufA, N);
  k_init_agg<<<(16 * N) / B256, B256, 0, stream>>>(bufA, dinv, b2, bufB, 16 * N);
  k_scatter<<<scatter_blocks, B256, 0, stream>>>(esrc, edst, dinv, bufA, bufB, E);

  // hf = relu(bufB) viewed as [64, 16384] (relu fused into split-K A loads)
  const int K = 16384, NSPLIT = 64, KLEN = K / NSPLIT;  // KLEN = 256

  k_zero<<<(64 * 32 + 64 * 16 + B256 - 1) / B256, B256, 0, stream>>>(pacc, 64 * 32 + 64 * 16);

  // policy hidden: 4 x 2 x 64 = 512 waves -> 64 blocks
  k_gemm_splitk_relu<KLEN><<<64, B256, 0, stream>>>(bufB, K, p1w, 32, pacc, 2, NSPLIT);
  // value hidden: 4 x 1 x 64 = 256 waves -> 32 blocks
  k_gemm_splitk_relu<KLEN><<<32, B256, 0, stream>>>(bufB, K, v1w, 16, vacc, 1, NSPLIT);

  k_bias_relu<<<(64 * 32) / B256, B256, 0, stream>>>(pacc, p1b, phid, 64 * 32, 32);
  k_bias_relu<<<(64 * 16 + B256 - 1) / B256, B256, 0, stream>>>(vacc, v1b, vhid, 64 * 16, 16);

  // X = phid @ p2_w + p2_b : 256 waves -> 32 blocks
  k_gemm_final<<<32, B256, 0, stream>>>(phid, p2w, p2b, Xout);
  // V
  k_value_final<<<1, 64, 0, stream>>>(vhid, v2w, v2b, Vout);
}